// ScaleSpaceDetector_47553877901627
// MI455X (gfx1250) — compile-verified
//
#include <hip/hip_runtime.h>
#include <cmath>
#include <cstdint>
#include <cstddef>

// ---------------- types / constants ----------------
typedef float v2f  __attribute__((ext_vector_type(2)));
typedef float v8f  __attribute__((ext_vector_type(8)));
typedef unsigned int u32x4 __attribute__((ext_vector_type(4)));
typedef int  i32x4 __attribute__((ext_vector_type(4)));
typedef int  i32x8 __attribute__((ext_vector_type(8)));

#define NOCT_MAX 6
#define NLVL 6
#define NRL 5
#define KFEAT 2000
#define SELCAP 2048
#define EPSV 1e-8f

#if __has_builtin(__builtin_amdgcn_wmma_f32_16x16x4_f32) && defined(__gfx1250__)
#define HAVE_WMMA_F32 1
#endif
#if __has_builtin(__builtin_amdgcn_tensor_load_to_lds) && defined(__gfx1250__)
#define HAVE_TDM 1
#endif

struct F5 { float v[NRL]; };
struct FinMeta {
  const float* r[NOCT_MAX];
  int Hs[NOCT_MAX];
  int off[NOCT_MAX + 1];
  int noct, B;
  float mi;
};

// ---------------- device helpers ----------------
__device__ __forceinline__ unsigned ordf(float f) {
  unsigned u = __float_as_uint(f);
  return (u & 0x80000000u) ? ~u : (u | 0x80000000u);
}
__device__ __forceinline__ float unordf(unsigned k) {
  unsigned u = (k & 0x80000000u) ? (k ^ 0x80000000u) : ~k;
  return __uint_as_float(u);
}
__device__ __forceinline__ int reflect_idx(int i, int n) {
  while (i < 0 || i >= n) { if (i < 0) i = -i; if (i >= n) i = 2 * n - 2 - i; }
  return i;
}

// ---------------- separable Gaussian blur (WMMA f32 + TDM tile feed) ----------------
// 32x32 output tile per 128-thread block. IN tile is (R x R), R = 32+2P, contiguous
// stride R in LDS (TDM layout). Row pass then column pass as 16x16 WMMA tiles:
// data fragment x banded-weight fragment, K stepped by 4, fully unrolled (P is a
// template constant so the weight-fragment array lives in VGPRs).
template <int P>
__global__ __launch_bounds__(128) void k_blur_t(const float* __restrict__ src,
                                                float* __restrict__ dst,
                                                int H, int W, float sigma) {
  constexpr int R  = 32 + 2 * P;
  constexpr int KS = 2 * P + 1;
  constexpr int KP = (16 + 2 * P + 3) & ~3;
  constexpr int NK = KP / 4;
  constexpr int INSZ = 64 * R + 64;       // 64 rows of slack coverage + pad

  __shared__ float IN[INSZ];
  __shared__ float MID_T[32 * 66];        // transposed row-blur result, stride 66
  __shared__ float gkpad[64];             // band-padded weights at offset +16

  const int tid  = threadIdx.x;
  const int lane = tid & 31;
  const int wave = tid >> 5;
  const int b    = blockIdx.z;
  const int x0   = blockIdx.x * 32;
  const int y0   = blockIdx.y * 32;
  const size_t bofs = (size_t)b * H * W;

  // normalized Gaussian weights into padded band table (single writer per slot)
  float wsum = 0.f;
  for (int i = 0; i < KS; ++i) { float xx = (float)(i - P); wsum += expf(-xx * xx / (2.f * sigma * sigma)); }
  if (tid < 64) {
    int d = tid - 16;
    float g = 0.f;
    if (d >= 0 && d < KS) { float xx = (float)(d - P); g = expf(-xx * xx / (2.f * sigma * sigma)) / wsum; }
    gkpad[tid] = g;
  }
  // zero slack beyond R*R so zero-weight WMMA reads never see NaN bits
  for (int i = R * R + tid; i < INSZ; i += 128) IN[i] = 0.f;

  const bool interior = (x0 - P >= 0) && (y0 - P >= 0) && (x0 - P + R <= W) && (y0 - P + R <= H);
  bool loaded = false;
#ifdef HAVE_TDM
  if (interior) {
    if (wave == 0) {
      // 2D Tensor DMA descriptor: 4-byte elems, tile R x R, row stride W
      unsigned long long ga = (unsigned long long)(uintptr_t)(src + bofs + (size_t)(y0 - P) * W + (x0 - P));
      unsigned ldsa = (unsigned)(uintptr_t)(&IN[0]);
      u32x4 g0;
      g0[0] = 1u;                                             // count=1
      g0[1] = ldsa;                                           // lds_addr
      g0[2] = (unsigned)(ga & 0xffffffffull);                 // global_addr lo
      g0[3] = ((unsigned)((ga >> 32) & 0x01ffffffull)) | 0x80000000u; // ga hi | type=2
      i32x8 g1;
      g1[0] = (int)(2u << 16);                                // data_size=4B, mask=0
      g1[1] = (int)((unsigned)(R & 0xffff) << 16);            // tensor_dim0 lo16
      g1[2] = (int)((unsigned)(R & 0xffff) << 16);            // tensor_dim1 lo16
      g1[3] = (int)((unsigned)(R & 0xffff) << 16);            // tile_dim0
      g1[4] = R;                                              // tile_dim1 (tile_dim2=0)
      g1[5] = W;                                              // tensor_dim0_stride lo32
      g1[6] = 0;
      g1[7] = 0;
      i32x4 gz4 = {0, 0, 0, 0};
      i32x8 gz8 = {0, 0, 0, 0, 0, 0, 0, 0};
      __builtin_amdgcn_tensor_load_to_lds(g0, g1, gz4, gz4, gz8, 0);
      __builtin_amdgcn_s_wait_tensorcnt(0);
    }
    loaded = true;
  }
#endif
  if (!loaded) {
    for (int i = tid; i < R * R; i += 128) {
      int rr = i / R, cc = i - rr * R;
      int yy = reflect_idx(y0 - P + rr, H);
      int xx = reflect_idx(x0 - P + cc, W);
      IN[i] = src[bofs + (size_t)yy * W + xx];
    }
  }
  __syncthreads();

#ifdef HAVE_WMMA_F32
  // Banded weight fragment: identical per-lane values for phase-2 B and phase-3 A:
  // w(k, lane) = g[k - (lane&15)], k = 4*kk + 2*(lane>=16) + {0,1}. Registers only.
  const int cl   = lane & 15;
  const int ksel = (lane >> 4) * 2;
  v2f wreg[NK];
#pragma unroll
  for (int kk = 0; kk < NK; ++kk) {
    int k = kk * 4 + ksel;
    wreg[kk].x = gkpad[16 + k - cl];
    wreg[kk].y = gkpad[16 + k + 1 - cl];
  }

  // ---- phase 2: row blur. MID[rr][c] = sum_j IN[rr][c+j]*g[j] (stored transposed) ----
  for (int t = wave; t < 8; t += 4) {
    const int R0 = (t & 3) * 16;
    const int C0 = (t >> 2) * 16;
    v8f acc = {0.f, 0.f, 0.f, 0.f, 0.f, 0.f, 0.f, 0.f};
    const int mrow = R0 + cl;
#pragma unroll
    for (int kk = 0; kk < NK; ++kk) {
      const int k = kk * 4 + ksel;
      v2f a;
      a.x = IN[mrow * R + C0 + k];
      a.y = IN[mrow * R + C0 + k + 1];
      acc = __builtin_amdgcn_wmma_f32_16x16x4_f32(false, a, false, wreg[kk], (short)0, acc, false, false);
    }
    const int mb = R0 + ((lane >> 4) ? 8 : 0);
#pragma unroll
    for (int v = 0; v < 8; ++v) MID_T[(C0 + cl) * 66 + mb + v] = acc[v];
  }
  __syncthreads();

  // ---- phase 3: column blur. OUT[y][x] = sum_j MID[y+j][x]*g[j] ----
  {
    const int Y0 = (wave & 1) * 16;
    const int X0 = (wave >> 1) * 16;
    v8f acc = {0.f, 0.f, 0.f, 0.f, 0.f, 0.f, 0.f, 0.f};
#pragma unroll
    for (int kk = 0; kk < NK; ++kk) {
      const int k = kk * 4 + ksel;
      v2f bm;
      bm.x = MID_T[(X0 + cl) * 66 + Y0 + k];
      bm.y = MID_T[(X0 + cl) * 66 + Y0 + k + 1];
      acc = __builtin_amdgcn_wmma_f32_16x16x4_f32(false, wreg[kk], false, bm, (short)0, acc, false, false);
    }
    const int mb = (lane >> 4) ? 8 : 0;
#pragma unroll
    for (int v = 0; v < 8; ++v) {
      int yy = y0 + Y0 + mb + v;
      int xx = x0 + X0 + cl;
      if (yy < H && xx < W) dst[bofs + (size_t)yy * W + xx] = acc[v];
    }
  }
#else
  for (int i = tid; i < 64 * 32; i += 128) {
    int rr = i >> 5, c = i & 31;
    float s = 0.f;
    if (rr < R) for (int j = 0; j < KS; ++j) s += IN[rr * R + c + j] * gkpad[16 + j];
    MID_T[c * 66 + rr] = s;
  }
  __syncthreads();
  for (int i = tid; i < 32 * 32; i += 128) {
    int yloc = i >> 5, xloc = i & 31;
    float s = 0.f;
    for (int j = 0; j < KS; ++j) s += MID_T[xloc * 66 + yloc + j] * gkpad[16 + j];
    int yy = y0 + yloc, xx = x0 + xloc;
    if (yy < H && xx < W) dst[bofs + (size_t)yy * W + xx] = s;
  }
#endif
}

// Generic (runtime-p) scalar fallback; unreachable with kornia's sigma set.
__global__ __launch_bounds__(128) void k_blur_generic(const float* __restrict__ src,
                                                      float* __restrict__ dst,
                                                      int H, int W, int p, float sigma) {
  __shared__ float IN[3840];
  __shared__ float MID[64 * 36];
  __shared__ float gkpad[64];
  const int tid = threadIdx.x;
  const int b = blockIdx.z;
  const int x0 = blockIdx.x * 32, y0 = blockIdx.y * 32;
  const int R = 32 + 2 * p, ks = 2 * p + 1;
  const size_t bofs = (size_t)b * H * W;
  float wsum = 0.f;
  for (int i = 0; i < ks; ++i) { float xx = (float)(i - p); wsum += expf(-xx * xx / (2.f * sigma * sigma)); }
  if (tid < 64) {
    int d = tid - 16; float g = 0.f;
    if (d >= 0 && d < ks) { float xx = (float)(d - p); g = expf(-xx * xx / (2.f * sigma * sigma)) / wsum; }
    gkpad[tid] = g;
  }
  for (int i = tid; i < R * R; i += 128) {
    int rr = i / R, cc = i - rr * R;
    IN[i] = src[bofs + (size_t)reflect_idx(y0 - p + rr, H) * W + reflect_idx(x0 - p + cc, W)];
  }
  __syncthreads();
  for (int i = tid; i < 64 * 32; i += 128) {
    int rr = i >> 5, c = i & 31;
    float s = 0.f;
    if (rr < R) for (int j = 0; j < ks; ++j) s += IN[rr * R + c + j] * gkpad[16 + j];
    MID[rr * 36 + c] = s;
  }
  __syncthreads();
  for (int i = tid; i < 32 * 32; i += 128) {
    int yloc = i >> 5, xloc = i & 31;
    float s = 0.f;
    for (int j = 0; j < ks; ++j) s += MID[(yloc + j) * 36 + xloc] * gkpad[16 + j];
    int yy = y0 + yloc, xx = x0 + xloc;
    if (yy < H && xx < W) dst[bofs + (size_t)yy * W + xx] = s;
  }
}

// ---------------- nearest-neighbor 2x downsample of level 3 ----------------
__global__ void k_down(const float* __restrict__ src, float* __restrict__ dst,
                       int Hd, int Wd) {
  int b = blockIdx.y;
  int i = blockIdx.x * 256 + threadIdx.x;
  int HWd = Hd * Wd;
  if (i >= HWd) return;
  int y = i / Wd, x = i - y * Wd;
  dst[(size_t)b * HWd + i] = src[(size_t)b * 4 * HWd + (size_t)(2 * y) * (2 * Wd) + 2 * x];
}

// ---------------- det(Hessian) * sigma^4, replicate padding ----------------
__global__ __launch_bounds__(256) void k_hessian(const float* __restrict__ lev,
                                                 float* __restrict__ r,
                                                 int B, int H, int W, F5 sig4) {
  int x = blockIdx.x * 16 + (threadIdx.x & 15);
  int y = blockIdx.y * 16 + (threadIdx.x >> 4);
  int z = blockIdx.z;               // z = l*B + b
  if (x >= W || y >= H) return;
  int l = z / B;
  const float* a = lev + (size_t)z * H * W;
  int ym = max(y - 1, 0), yp = min(y + 1, H - 1);
  int xm = max(x - 1, 0), xp = min(x + 1, W - 1);
  float m00 = a[(size_t)ym * W + xm], m01 = a[(size_t)ym * W + x], m02 = a[(size_t)ym * W + xp];
  float m10 = a[(size_t)y  * W + xm], m11 = a[(size_t)y  * W + x], m12 = a[(size_t)y  * W + xp];
  float m20 = a[(size_t)yp * W + xm], m21 = a[(size_t)yp * W + x], m22 = a[(size_t)yp * W + xp];
  float gxx = (m00 - 2.f * m01 + m02 + 2.f * m10 - 4.f * m11 + 2.f * m12 + m20 - 2.f * m21 + m22) * (1.f / 16.f);
  float gyy = (m00 + 2.f * m01 + m02 - 2.f * m10 - 4.f * m11 - 2.f * m12 + m20 + 2.f * m21 + m22) * (1.f / 16.f);
  float gxy = (-m00 + m02 + m20 - m22) * 0.25f;
  r[(size_t)z * H * W + (size_t)y * W + x] = (gxx * gyy - gxy * gxy) * sig4.v[l];
}

// ---------------- per-(batch,octave) max of r (orderable-uint atomics) ----------------
__global__ __launch_bounds__(256) void k_rmax(const float* __restrict__ r,
                                              unsigned* __restrict__ mx,
                                              int B, int HW, int oct) {
  __shared__ float red[256];
  int b = blockIdx.y;
  int n = NRL * HW;
  float m = -3.4e38f;
  for (int i = blockIdx.x * blockDim.x + threadIdx.x; i < n; i += blockDim.x * gridDim.x) {
    int l = i / HW, j = i - l * HW;
    m = fmaxf(m, r[((size_t)l * B + b) * HW + j]);
  }
  red[threadIdx.x] = m;
  __syncthreads();
  for (int s = 128; s > 0; s >>= 1) {
    if (threadIdx.x < s) red[threadIdx.x] = fmaxf(red[threadIdx.x], red[threadIdx.x + s]);
    __syncthreads();
  }
  if (threadIdx.x == 0) atomicMax(&mx[b * NOCT_MAX + oct], ordf(red[0]));
}

// ---------------- soft-pooled value per candidate ----------------
__global__ __launch_bounds__(256) void k_softval(const float* __restrict__ r,
                                                 const unsigned* __restrict__ mx,
                                                 float* __restrict__ valout,
                                                 int B, int H, int W, int oct,
                                                 int coff, int C) {
  __shared__ float rt[5 * 18 * 18];
  __shared__ float et[5 * 18 * 18];
  const int b = blockIdx.z;
  const int x0 = blockIdx.x * 16, y0 = blockIdx.y * 16;
  const float mxf = unordf(mx[b * NOCT_MAX + oct]);
  const size_t HW = (size_t)H * W;
  for (int i = threadIdx.x; i < 5 * 18 * 18; i += 256) {
    int l = i / 324; int rem = i - l * 324; int yy = rem / 18; int xx = rem - yy * 18;
    int gy = y0 - 1 + yy, gx = x0 - 1 + xx;
    float rv = 0.f, ev = 0.f;
    if (gy >= 0 && gy < H && gx >= 0 && gx < W) {
      rv = r[((size_t)l * B + b) * HW + (size_t)gy * W + gx];
      ev = expf(rv - mxf);
    }
    rt[i] = rv; et[i] = ev;
  }
  __syncthreads();
  int xl = threadIdx.x & 15, yl = threadIdx.x >> 4;
  int gx = x0 + xl, gy = y0 + yl;
  if (gx >= W || gy >= H) return;
  for (int l = 0; l < NRL; ++l) {
    float S = 0.f, V = 0.f;
    for (int dz = -1; dz <= 1; ++dz) {
      int z = l + dz; if (z < 0 || z >= NRL) continue;
      for (int dy = 0; dy <= 2; ++dy)
        for (int dx = 0; dx <= 2; ++dx) {
          int ii = z * 324 + (yl + dy) * 18 + (xl + dx);
          float e = et[ii];
          S += e; V += e * rt[ii];
        }
    }
    valout[(size_t)b * C + coff + ((size_t)l * H + gy) * W + gx] = V / (S + EPSV);
  }
}

// ---------------- selection: radix-select + gather + bitonic sort ----------------
__global__ void k_zero_u32(unsigned* p, int n) {
  int i = blockIdx.x * 256 + threadIdx.x;
  if (i < n) p[i] = 0u;
}
__global__ void k_init_sel(unsigned long long* sel, unsigned* cnt, unsigned* state, int B) {
  int i = blockIdx.x * 256 + threadIdx.x;
  if (i < B * SELCAP) sel[i] = 0ull;
  if (i < B) {
    cnt[i] = 0u;
    state[i * 4 + 0] = 0u; state[i * 4 + 1] = KFEAT;
    state[i * 4 + 2] = 0u; state[i * 4 + 3] = 0u;
  }
}
__global__ __launch_bounds__(256) void k_hist(const float* __restrict__ val,
                                              const unsigned* __restrict__ state,
                                              unsigned* __restrict__ hist, int C, int rd) {
  __shared__ unsigned lh[256];
  int b = blockIdx.y;
  for (int i = threadIdx.x; i < 256; i += 256) lh[i] = 0u;
  __syncthreads();
  unsigned prefix = state[b * 4 + 0];
  int shift = (3 - rd) * 8;
  for (int i = blockIdx.x * 256 + threadIdx.x; i < C; i += 256 * gridDim.x) {
    unsigned key = ordf(val[(size_t)b * C + i]);
    if (rd == 0 || (key >> (shift + 8)) == prefix)
      atomicAdd(&lh[(key >> shift) & 0xffu], 1u);
  }
  __syncthreads();
  for (int i = threadIdx.x; i < 256; i += 256)
    if (lh[i]) atomicAdd(&hist[b * 256 + i], lh[i]);
}
__global__ void k_scanbin(unsigned* state, const unsigned* hist, int rd) {
  int b = blockIdx.x;
  if (threadIdx.x != 0) return;
  unsigned k = state[b * 4 + 1];
  const unsigned* h = &hist[b * 256];
  unsigned cum = 0; int sel = 0;
  for (int bin = 255; bin >= 0; --bin) {
    unsigned c = h[bin];
    if (cum + c >= k) { sel = bin; k = k - cum; break; }
    cum += c;
  }
  state[b * 4 + 0] = (state[b * 4 + 0] << 8) | (unsigned)sel;
  state[b * 4 + 1] = k;
  if (rd == 3) state[b * 4 + 2] = state[b * 4 + 0];
}
__global__ __launch_bounds__(256) void k_gather(const float* __restrict__ val,
                                                const unsigned* __restrict__ state,
                                                unsigned* __restrict__ cnt,
                                                unsigned long long* __restrict__ sel, int C) {
  int b = blockIdx.y;
  unsigned T = state[b * 4 + 2];
  for (int i = blockIdx.x * 256 + threadIdx.x; i < C; i += 256 * gridDim.x) {
    unsigned key = ordf(val[(size_t)b * C + i]);
    if (key >= T) {
      unsigned pos = atomicAdd(&cnt[b], 1u);
      if (pos < SELCAP)
        sel[(size_t)b * SELCAP + pos] = ((unsigned long long)key << 32) | (unsigned)(~(unsigned)i);
    }
  }
}
__global__ __launch_bounds__(256) void k_sort(unsigned long long* sel) {
  __shared__ unsigned long long key[SELCAP];
  unsigned long long* s = sel + (size_t)blockIdx.x * SELCAP;
  for (int i = threadIdx.x; i < SELCAP; i += 256) key[i] = s[i];
  __syncthreads();
  for (unsigned size = 2; size <= SELCAP; size <<= 1) {
    for (unsigned stride = size >> 1; stride > 0; stride >>= 1) {
      for (unsigned i = threadIdx.x; i < SELCAP / 2; i += 256) {
        unsigned pos = 2 * i - (i & (stride - 1));
        unsigned long long ka = key[pos], kb = key[pos + stride];
        bool desc = ((pos & size) == 0);
        bool sw = desc ? (ka < kb) : (ka > kb);
        if (sw) { key[pos] = kb; key[pos + stride] = ka; }
      }
      __syncthreads();
    }
  }
  for (int i = threadIdx.x; i < SELCAP; i += 256) s[i] = key[i];
}

// ---------------- finalize: soft-argmax coords + LAF build ----------------
__global__ __launch_bounds__(256) void k_finalize(const unsigned long long* __restrict__ sel,
                                                  const unsigned* __restrict__ mx,
                                                  float* __restrict__ out, FinMeta fm) {
  int i = blockIdx.x * 256 + threadIdx.x;
  int b = blockIdx.y;
  if (i >= KFEAT) return;
  int B = fm.B;
  float* laf  = out + (size_t)(b * KFEAT + i) * 6;
  float* resp = out + (size_t)B * KFEAT * 6 + (size_t)b * KFEAT + i;
  unsigned long long k64 = sel[(size_t)b * SELCAP + i];
  if (k64 == 0ull) {
    for (int j = 0; j < 6; ++j) laf[j] = 0.f;
    *resp = 0.f;
    return;
  }
  int idx = (int)(~(unsigned)(k64 & 0xffffffffull));
  int o = 0;
  while (o + 1 < fm.noct && idx >= fm.off[o + 1]) ++o;
  int rem = idx - fm.off[o];
  int H = fm.Hs[o], W = H, HW = H * W;
  int l = rem / HW; rem -= l * HW;
  int y = rem / W; int x = rem - y * W;
  const float* r = fm.r[o];
  float mxf = unordf(mx[b * NOCT_MAX + o]);
  float S = 0.f, V = 0.f, Sz = 0.f, Sy = 0.f, Sx = 0.f;
  for (int dz = -1; dz <= 1; ++dz) {
    int z = l + dz; if (z < 0 || z >= NRL) continue;
    for (int dy = -1; dy <= 1; ++dy) {
      int yy = y + dy;
      for (int dx = -1; dx <= 1; ++dx) {
        int xx = x + dx;
        float e = 0.f, rv = 0.f;
        if (yy >= 0 && yy < H && xx >= 0 && xx < W) {
          rv = r[((size_t)z * B + b) * HW + (size_t)yy * W + xx];
          e = expf(rv - mxf);
        }
        S += e; V += e * rv;
        Sz += e * (float)z; Sy += e * (float)yy; Sx += e * (float)xx;
      }
    }
  }
  float den = S + EPSV;
  float cz = (float)l + (Sz - (float)l * S) / den;
  float cy = (float)y + (Sy - (float)y * S) / den;
  float cx = (float)x + (Sx - (float)x * S) / den;
  float s = 9.6f * exp2f(cz * (1.f / 3.f));          // MR * INIT_SIGMA * 2^(cz/3)
  float m = (float)H;                                 // min(Ho,Wo)
  float px = (float)(1 << o);
  float a = s / m * fm.mi;                            // denormalize_laf coef
  laf[0] = a;   laf[1] = 0.f; laf[2] = cx * px;
  laf[3] = 0.f; laf[4] = a;   laf[5] = cy * px;
  *resp = V / den;
}

// ---------------- host orchestration ----------------
static void launch_blur(const float* src, float* dst, int Ho, int Wo, int B,
                        int p, float sigma, hipStream_t stream) {
  dim3 bg((Wo + 31) / 32, (Ho + 31) / 32, B);
  switch (p) {
    case 5:  k_blur_t<5><<<bg, 128, 0, stream>>>(src, dst, Ho, Wo, sigma); break;
    case 6:  k_blur_t<6><<<bg, 128, 0, stream>>>(src, dst, Ho, Wo, sigma); break;
    case 8:  k_blur_t<8><<<bg, 128, 0, stream>>>(src, dst, Ho, Wo, sigma); break;
    case 10: k_blur_t<10><<<bg, 128, 0, stream>>>(src, dst, Ho, Wo, sigma); break;
    case 12: k_blur_t<12><<<bg, 128, 0, stream>>>(src, dst, Ho, Wo, sigma); break;
    default: k_blur_generic<<<bg, 128, 0, stream>>>(src, dst, Ho, Wo, p, sigma); break;
  }
}

extern "C" void kernel_launch(void* const* d_in, const int* in_sizes, int n_in,
                              void* d_out, int out_size, void* d_ws, size_t ws_size,
                              hipStream_t stream) {
  (void)n_in; (void)out_size; (void)ws_size;
  const float* img = (const float*)d_in[0];
  const int H0 = 512, W0 = 512;
  const int B = in_sizes[0] / (H0 * W0);

  // octave sizes (stop when next octave min dim <= 15)
  int Hs[NOCT_MAX]; int noct = 0;
  { int h = H0;
    for (;;) { Hs[noct++] = h; int n2 = h / 2; if (n2 <= 15 || noct >= NOCT_MAX) break; h = n2; } }

  // sigmas (kornia ScalePyramid defaults)
  const double step = pow(2.0, 1.0 / 3.0);
  double lev_sig[NLVL], blur_sig[NRL];
  for (int i = 0; i < NLVL; ++i) lev_sig[i] = 1.6 * pow(step, (double)i);
  for (int i = 0; i < NRL; ++i) blur_sig[i] = lev_sig[i] * sqrt(step * step - 1.0);
  double sigma0 = sqrt(1.6 * 1.6 - 0.25);
  F5 sig4; for (int i = 0; i < NRL; ++i) sig4.v[i] = (float)pow(lev_sig[i], 4.0);
  auto halfk = [](double s) { int k = (int)(2.0 * 4.0 * s + 1.0); if ((k & 1) == 0) k++; return k / 2; };

  // workspace layout
  float* wsf = (float*)d_ws;
  size_t off = 0;
  size_t lev_off[NOCT_MAX], r_off_[NOCT_MAX];
  int coff[NOCT_MAX + 1]; coff[0] = 0;
  for (int o = 0; o < noct; ++o) {
    size_t HW = (size_t)Hs[o] * Hs[o];
    lev_off[o] = off; off += (size_t)NLVL * B * HW;
    r_off_[o]  = off; off += (size_t)NRL * B * HW;
    coff[o + 1] = coff[o] + NRL * (int)HW;
  }
  const int C = coff[noct];
  size_t val_off = off; off += (size_t)B * C;
  size_t ubase = (off * sizeof(float) + 255) & ~(size_t)255;
  unsigned* mx    = (unsigned*)((char*)d_ws + ubase);   // B*NOCT_MAX
  unsigned* hist  = mx + B * NOCT_MAX;                  // B*256
  unsigned* state = hist + B * 256;                     // B*4
  unsigned* cnt   = state + B * 4;                      // B
  size_t selb = (((size_t)((char*)(cnt + B) - (char*)d_ws)) + 255) & ~(size_t)255;
  unsigned long long* sel = (unsigned long long*)((char*)d_ws + selb);

  k_zero_u32<<<(B * NOCT_MAX + 255) / 256, 256, 0, stream>>>(mx, B * NOCT_MAX);

  for (int o = 0; o < noct; ++o) {
    int Ho = Hs[o], Wo = Ho;
    size_t HW = (size_t)Ho * Wo;
    float* lev = wsf + lev_off[o];
    float* rr  = wsf + r_off_[o];
    if (o == 0) {
      launch_blur(img, lev, Ho, Wo, B, halfk(sigma0), (float)sigma0, stream);
    } else {
      const float* src3 = wsf + lev_off[o - 1] + (size_t)3 * B * Hs[o - 1] * Hs[o - 1];
      k_down<<<dim3((unsigned)((HW + 255) / 256), B), 256, 0, stream>>>(src3, lev, Ho, Wo);
    }
    for (int i = 0; i < NRL; ++i) {
      const float* s = lev + (size_t)i * B * HW;
      float* d = lev + (size_t)(i + 1) * B * HW;
      launch_blur(s, d, Ho, Wo, B, halfk(blur_sig[i]), (float)blur_sig[i], stream);
    }
    dim3 hg((Wo + 15) / 16, (Ho + 15) / 16, NRL * B);
    k_hessian<<<hg, 256, 0, stream>>>(lev, rr, B, Ho, Wo, sig4);
    int mgx = (int)((NRL * HW + 255) / 256); if (mgx > 64) mgx = 64; if (mgx < 1) mgx = 1;
    k_rmax<<<dim3(mgx, B), 256, 0, stream>>>(rr, mx, B, (int)HW, o);
    dim3 sg((Wo + 15) / 16, (Ho + 15) / 16, B);
    k_softval<<<sg, 256, 0, stream>>>(rr, mx, wsf + val_off, B, Ho, Wo, o, coff[o], C);
  }

  // global top-2000 per image: 4-round radix-select, gather, bitonic sort
  k_init_sel<<<(B * SELCAP + 255) / 256, 256, 0, stream>>>(sel, cnt, state, B);
  for (int rd = 0; rd < 4; ++rd) {
    k_zero_u32<<<(B * 256 + 255) / 256, 256, 0, stream>>>(hist, B * 256);
    k_hist<<<dim3(256, B), 256, 0, stream>>>(wsf + val_off, state, hist, C, rd);
    k_scanbin<<<B, 32, 0, stream>>>(state, hist, rd);
  }
  k_gather<<<dim3(256, B), 256, 0, stream>>>(wsf + val_off, state, cnt, sel, C);
  k_sort<<<B, 256, 0, stream>>>(sel);

  FinMeta fm;
  for (int o = 0; o < NOCT_MAX; ++o) {
    int oo = (o < noct) ? o : (noct - 1);
    fm.r[o] = wsf + r_off_[oo];
    fm.Hs[o] = Hs[oo];
  }
  for (int o = 0; o <= noct; ++o) fm.off[o] = coff[o];
  fm.noct = noct; fm.B = B;
  fm.mi = (float)((H0 < W0) ? H0 : W0);
  k_finalize<<<dim3((KFEAT + 255) / 256, B), 256, 0, stream>>>(sel, mx, (float*)d_out, fm);
}